// SphereHashGridBackground_47605417508928
// MI455X (gfx1250) — compile-verified
//
#include <hip/hip_runtime.h>
#include <math.h>

// ---------------------------------------------------------------------------
// SphereHashGridBackground: ray-sphere intersection + 6-level NGP hash grid.
// Gather-bound; table (50 MB) is L2-resident on MI455X (192 MB L2), so the
// floor is HBM streaming of inputs/outputs (~176 MB -> ~7.5 us @ 23.3 TB/s).
// CDNA5 paths: 16x straight-line global_load_async_to_lds_b128 (ASYNCcnt)
// staging the 64 KB level-0 dense table, ds_load_b128 gathers from LDS,
// s_wait_asynccnt, b128 global gathers/stores everywhere else.
// ---------------------------------------------------------------------------

namespace {
constexpr int      kNumLevels   = 6;
constexpr int      kLevelDim    = 4;
constexpr uint32_t kHashmapSize = 1u << 19;
constexpr float    kRadius      = 500.0f;
constexpr int      kBlock       = 256;
constexpr int      kL0Entries   = 16 * 16 * 16;   // 4096 float4 = 64 KB
constexpr int      kChunks      = kL0Entries / kBlock;   // 16
constexpr int      kChunkBytes  = kBlock * 16;           // 4096 B per step
}

#if __has_builtin(__builtin_amdgcn_global_load_async_to_lds_b128)
#define HGE_ASYNC_LDS 1
#else
#define HGE_ASYNC_LDS 0
#endif

typedef int hge_v4i __attribute__((vector_size(16)));
typedef hge_v4i __attribute__((address_space(1))) hge_gv4i;   // global
typedef hge_v4i __attribute__((address_space(3))) hge_sv4i;   // LDS

__global__ __launch_bounds__(kBlock) void hash_grid_bg_kernel(
    const float* __restrict__ dirs, const float* __restrict__ origs,
    const float* __restrict__ table, float* __restrict__ out, int nrays)
{
    __shared__ float4 lds0[kL0Entries];   // level-0 dense table, 64 KB

    const float4* t0 = reinterpret_cast<const float4*>(table);  // level 0 @ 0
    const int tid = threadIdx.x;

#if HGE_ASYNC_LDS
    // 16 straight-line async copies; the 24-bit INST_OFFSET is applied to
    // both the global source and the LDS destination (same 4096 B stride).
    {
        hge_gv4i* gsrc = (hge_gv4i*)(t0 + tid);
        hge_sv4i* ldst = (hge_sv4i*)(&lds0[tid]);
#define HGE_ASYNC_CHUNK(K) \
        __builtin_amdgcn_global_load_async_to_lds_b128(gsrc, ldst, (K) * kChunkBytes, 0)
        HGE_ASYNC_CHUNK(0);  HGE_ASYNC_CHUNK(1);  HGE_ASYNC_CHUNK(2);  HGE_ASYNC_CHUNK(3);
        HGE_ASYNC_CHUNK(4);  HGE_ASYNC_CHUNK(5);  HGE_ASYNC_CHUNK(6);  HGE_ASYNC_CHUNK(7);
        HGE_ASYNC_CHUNK(8);  HGE_ASYNC_CHUNK(9);  HGE_ASYNC_CHUNK(10); HGE_ASYNC_CHUNK(11);
        HGE_ASYNC_CHUNK(12); HGE_ASYNC_CHUNK(13); HGE_ASYNC_CHUNK(14); HGE_ASYNC_CHUNK(15);
#undef HGE_ASYNC_CHUNK
    }
#if __has_builtin(__builtin_amdgcn_s_wait_asynccnt)
    __builtin_amdgcn_s_wait_asynccnt(0);
#else
    asm volatile("s_wait_asynccnt 0x0" ::: "memory");
#endif
#else
    for (int i = tid; i < kL0Entries; i += kBlock)
        lds0[i] = t0[i];
#endif
    __syncthreads();

    const int n = blockIdx.x * kBlock + tid;
    if (n >= nrays) return;

    // ---- ray-sphere intersection (a == 1, dirs normalized) ----
    const float dx = dirs[3*n+0], dy = dirs[3*n+1], dz = dirs[3*n+2];
    const float ox = origs[3*n+0], oy = origs[3*n+1], oz = origs[3*n+2];

    const float b    = 2.0f * (ox*dx + oy*dy + oz*dz);
    const float c    = (ox*ox + oy*oy + oz*oz) - kRadius * kRadius;
    const float disc = b*b - 4.0f*c;
    const float t    = (-b + sqrtf(fmaxf(disc, 0.0f))) * 0.5f;

    const float inv2R = 0.5f / kRadius;
    const float cx = fminf(fmaxf((ox + t*dx + kRadius) * inv2R, 0.0f), 1.0f);
    const float cy = fminf(fmaxf((oy + t*dy + kRadius) * inv2R, 0.0f), 1.0f);
    const float cz = fminf(fmaxf((oz + t*dz + kRadius) * inv2R, 0.0f), 1.0f);

    float* outp = out + (size_t)n * (kNumLevels * kLevelDim);

#pragma unroll
    for (int L = 0; L < kNumLevels; ++L) {
        const uint32_t res   = 16u << L;                // 16,32,64,128,256,512
        const float    scale = (float)res - 1.0f;       // base*2^L - 1 exactly
        const bool     dense = (res <= 64u);            // res^3 <= 2^19

        const float px = cx*scale + 0.5f; const float gx = floorf(px); const float fx = px - gx;
        const float py = cy*scale + 0.5f; const float gy = floorf(py); const float fy = py - gy;
        const float pz = cz*scale + 0.5f; const float gz = floorf(pz); const float fz = pz - gz;
        const uint32_t ix = (uint32_t)gx, iy = (uint32_t)gy, iz = (uint32_t)gz;

        const float wx[2] = {1.0f - fx, fx};
        const float wy[2] = {1.0f - fy, fy};
        const float wz[2] = {1.0f - fz, fz};

        const float4* tl = reinterpret_cast<const float4*>(table)
                           + (size_t)L * kHashmapSize;

        float a0 = 0.f, a1 = 0.f, a2 = 0.f, a3 = 0.f;
#pragma unroll
        for (int corner = 0; corner < 8; ++corner) {
            const uint32_t ci = (corner >> 2) & 1u;
            const uint32_t cj = (corner >> 1) & 1u;
            const uint32_t ck =  corner       & 1u;
            uint32_t x = ix + ci, y = iy + cj, z = iz + ck;
            uint32_t idx;
            if (dense) {
                x = min(x, res - 1u); y = min(y, res - 1u); z = min(z, res - 1u);
                idx = x + y * res + z * res * res;
            } else {
                idx = (x ^ (y * 2654435761u) ^ (z * 805459861u))
                      & (kHashmapSize - 1u);
            }
            const float w = wx[ci] * wy[cj] * wz[ck];
            const float4 f = (L == 0) ? lds0[idx] : tl[idx];  // ds_load_b128 / global_load_b128
            a0 += f.x * w; a1 += f.y * w; a2 += f.z * w; a3 += f.w * w;
        }
        float4 r; r.x = a0; r.y = a1; r.z = a2; r.w = a3;
        *reinterpret_cast<float4*>(outp + L * kLevelDim) = r;   // global_store_b128
    }
}

extern "C" void kernel_launch(void* const* d_in, const int* in_sizes, int n_in,
                              void* d_out, int out_size, void* d_ws, size_t ws_size,
                              hipStream_t stream) {
    const float* dirs  = (const float*)d_in[0];   // view_dirs  [N,3]
    const float* origs = (const float*)d_in[1];   // ray_origins[N,3]
    const float* table = (const float*)d_in[2];   // [6, 2^19, 4]
    float*       out   = (float*)d_out;           // [N, 24]

    const int nrays  = in_sizes[0] / 3;
    const int blocks = (nrays + kBlock - 1) / kBlock;
    hipLaunchKernelGGL(hash_grid_bg_kernel, dim3(blocks), dim3(kBlock), 0, stream,
                       dirs, origs, table, out, nrays);
}